// MaxUnpooling2D_90074054132394
// MI455X (gfx1250) — compile-verified
//
#include <hip/hip_runtime.h>

// MaxUnpooling2D scatter-add for MI455X (gfx1250).
// Shapes fixed by the reference:
//   updates: (8,128,128,256) f32  -> N_IN  = 33,554,432
//   mask:    same, int32, values in [0, 16,777,216)
//   out:     (8,256,256,256) f32  -> N_OUT = 134,217,728
// flat_input_size = 128*128*256 = 1<<22, so batch offset = i & ~((1<<22)-1).

static constexpr int BATCH_MASK = ~((1 << 22) - 1);  // 0xFFC00000

// ---- Kernel 1: zero-fill output with 128-bit stores -------------------------
__global__ void unpool_zero_kernel(float4* __restrict__ out4, int n4) {
    int i = blockIdx.x * blockDim.x + threadIdx.x;
    if (i < n4) {
        out4[i] = make_float4(0.f, 0.f, 0.f, 0.f);
    }
}

// ---- Kernel 2: scatter-add, 4 elements per lane -----------------------------
__global__ void unpool_scatter_kernel(const float* __restrict__ upd,
                                      const int* __restrict__ mask,
                                      float* __restrict__ out,
                                      int n) {
    const int i = (blockIdx.x * blockDim.x + threadIdx.x) * 4;
    if (i >= n) return;

    // Prefetch the streaming inputs a few KB ahead (global_prefetch_b8,
    // fire-and-forget: no LOADcnt, dropped if OOB-speculative).
    __builtin_prefetch(upd + i + 4096, 0, 0);
    __builtin_prefetch(mask + i + 4096, 0, 0);

    // 128-bit coalesced loads (global_load_b128).
    const float4 u = *reinterpret_cast<const float4*>(upd + i);
    const int4   m = *reinterpret_cast<const int4*>(mask + i);

    // batch * flat_input_size == (i >> 22) << 22 == i & BATCH_MASK,
    // identical for all 4 packed elements (i is 4-aligned).
    const int base = i & BATCH_MASK;

    // Non-returning f32 atomics -> global_atomic_add_f32 (STOREcnt-tracked).
    atomicAdd(out + (m.x + base), u.x);
    atomicAdd(out + (m.y + base), u.y);
    atomicAdd(out + (m.z + base), u.z);
    atomicAdd(out + (m.w + base), u.w);
}

extern "C" void kernel_launch(void* const* d_in, const int* in_sizes, int n_in,
                              void* d_out, int out_size, void* d_ws, size_t ws_size,
                              hipStream_t stream) {
    const float* upd  = (const float*)d_in[0];
    const int*   mask = (const int*)d_in[1];
    float*       out  = (float*)d_out;
    const int n = in_sizes[0];          // 33,554,432

    // 1) Zero the 512 MB output with B128 stores.
    const int n4 = out_size / 4;        // out_size is a multiple of 4
    {
        const int threads = 256;
        const int blocks  = (n4 + threads - 1) / threads;
        unpool_zero_kernel<<<blocks, threads, 0, stream>>>((float4*)out, n4);
    }

    // 2) Scatter-add (stream-ordered after the zero fill).
    {
        const int threads = 256;
        const int quads   = n / 4;      // n is a multiple of 4
        const int blocks  = (quads + threads - 1) / threads;
        unpool_scatter_kernel<<<blocks, threads, 0, stream>>>(upd, mask, out, n);
    }
}